// Conv_Capsule_15315853377613
// MI455X (gfx1250) — compile-verified
//
#include <hip/hip_runtime.h>

typedef __attribute__((ext_vector_type(2))) float v2f;
typedef __attribute__((ext_vector_type(8))) float v8f;
typedef int v4i_ __attribute__((vector_size(16)));   // matches builtin param type

#define KH 3
#define KW 3
#define IC 8
#define IDIM 8
#define OC 16
#define OCP 17              // pad so per-lane ds_add on b hits 8 distinct banks
#define DV 16
#define NI (KH*KW*IC)       // 72
#define HP 11
#define WP 11
#define PPW 16              // positions per workgroup == WMMA N (transposed)
#define NPOS (32*HP*WP)     // 3872
#define NROUT 3
#define NT 512              // 16 wave32 waves; wave id == output capsule
#define CAPS_EPS 1e-7f
#define WP_ELEMS (NI*OC*32) // packed-W float4 count (589824 bytes)

#if defined(__HIP_DEVICE_COMPILE__) && \
    __has_builtin(__builtin_amdgcn_global_load_async_to_lds_b128) && \
    __has_builtin(__builtin_amdgcn_s_wait_asynccnt)
#define USE_ASYNC_LDS 1
#else
#define USE_ASYNC_LDS 0
#endif

// One-time repack of W into WMMA A-fragment order: Wp[i][o][lane] = float4 of
// exactly the 4 elements lane needs ({p=2h,2h+1,2h+4,2h+5} at d=nlo). Turns 4
// scattered global_load_b32 per (i,o) tile into ONE coalesced global_load_b128.
__global__ __launch_bounds__(256) void repack_w_kernel(const float* __restrict__ Wg,
                                                       float4* __restrict__ Wp) {
    int idx = blockIdx.x * 256 + threadIdx.x;
    if (idx >= WP_ELEMS) return;
    int lane = idx & 31;
    int o    = (idx >> 5) & (OC - 1);
    int ii   = idx >> 9;
    int half = lane >> 4;
    int nlo  = lane & 15;
    const float* wb = Wg + ((ii * OC + o) * IDIM) * DV + nlo;
    float4 v;
    v.x = wb[(2 * half + 0) * DV];
    v.y = wb[(2 * half + 1) * DV];
    v.z = wb[(2 * half + 4) * DV];
    v.w = wb[(2 * half + 5) * DV];
    Wp[idx] = v;
}

template <bool PACKED>
__launch_bounds__(NT)
__global__ void caps_conv_route_kernel(const float* __restrict__ x,
                                       const float* __restrict__ Wg,
                                       const float4* __restrict__ Wp,
                                       const float* __restrict__ bias,
                                       float* __restrict__ out)
{
    __shared__ float patch[PPW][NI][IDIM];   // 36 KB   conv patches
    __shared__ float bL[PPW][NI][OCP];       // 76.5 KB routing logits
    __shared__ float cL[NI][OC][PPW];        // 72 KB   softmax coefficients

    const int tid  = threadIdx.x;
    const int o    = tid >> 5;       // wave == output capsule (16 waves)
    const int lane = tid & 31;
    const int half = lane >> 4;      // K-half selector / d-half selector
    const int nlo  = lane & 15;      // transposed layout: lane -> position
    const int pbase = blockIdx.x * PPW;

    // Warm GL2 with the heavily-reused weight tensor (global_prefetch_b8).
    const float* wpf = (const float*)Wp;
    for (int idx = tid * 32; idx < NI * OC * IDIM * DV; idx += NT * 32)
        __builtin_prefetch(PACKED ? (wpf + idx) : (Wg + idx), 0, 1);

    // Gather strided conv patches (contiguous float4 granules) into LDS,
    // via the gfx1250 async global->LDS path when available.
    for (int idx = tid; idx < PPW * NI * IDIM / 4; idx += NT) {
        int q   = idx & 1;                 // which half of the 8-wide pose dim
        int ii  = (idx >> 1) % NI;
        int m   = idx / (2 * NI);
        int P   = pbase + m;
        int bb  = P / (HP * WP);
        int rem = P - bb * (HP * WP);
        int xo  = rem / WP;
        int yo  = rem - xo * WP;
        int ki  = ii / (KW * IC);
        int r2  = ii - ki * (KW * IC);
        int kj  = r2 >> 3;
        int ic  = r2 & (IC - 1);
        int gx  = xo * 2 + ki;
        int gy  = yo * 2 + kj;
        const float* src = x + (((bb * 24 + gx) * 24 + gy) * IC + ic) * IDIM + 4 * q;
        float* dst = &patch[m][ii][4 * q];
#if USE_ASYNC_LDS
        __builtin_amdgcn_global_load_async_to_lds_b128(
            (__attribute__((address_space(1))) v4i_*)(void*)src,
            (__attribute__((address_space(3))) v4i_*)(void*)dst, 0, 0);
#else
        *(float4*)dst = *(const float4*)src;
#endif
    }
#if USE_ASYNC_LDS
    __builtin_amdgcn_s_wait_asynccnt(0);
#endif

    // Initialize routing logits from bias (broadcast over positions).
    for (int idx = tid; idx < PPW * NI * OC; idx += NT) {
        int oo = idx & (OC - 1);
        int ii = (idx >> 4) % NI;
        int m  = idx / (NI * OC);
        bL[m][ii][oo] = bias[ii * OC + oo];
    }
    __syncthreads();

    const v8f vz = {0.f, 0.f, 0.f, 0.f, 0.f, 0.f, 0.f, 0.f};
    v8f outf = vz;   // out^T fragment: lane -> pos, element j -> d = 8*half + j
    const float*  wbase  = Wg + (o * IDIM) * DV + nlo;      // unpacked path
    const float4* wpbase = Wp + (o * 32 + lane);            // packed path
    const int wstride = OC * IDIM * DV;

    for (int r = 0; r < NROUT; ++r) {
        // ---- Phase A: c = softmax over o of b  -> cL[i][o][pos]
        for (int row = tid; row < PPW * NI; row += NT) {
            int m  = row & (PPW - 1);
            int ii = row >> 4;
            float tv[OC];
            float mx = -3.0e38f;
            #pragma unroll
            for (int k = 0; k < OC; ++k) { tv[k] = bL[m][ii][k]; mx = fmaxf(mx, tv[k]); }
            float sum = 0.f;
            #pragma unroll
            for (int k = 0; k < OC; ++k) { tv[k] = __expf(tv[k] - mx); sum += tv[k]; }
            float inv = __frcp_rn(sum);
            #pragma unroll
            for (int k = 0; k < OC; ++k) cL[ii][k][m] = tv[k] * inv;
        }
        __syncthreads();

        // ---- Phase B: votes^T = W^T x patch^T via WMMA (operands swapped =
        // free transpose), s^T[d,pos] = sum_i c[pos,i,o] * votes^T[d,pos].
        v8f s = vz;
        for (int ii = 0; ii < NI; ++ii) {
            v2f pb0, pb1;                    // B frag = patch^T: lane -> pos
            pb0.x = patch[nlo][ii][2 * half + 0];
            pb0.y = patch[nlo][ii][2 * half + 1];
            pb1.x = patch[nlo][ii][2 * half + 4];
            pb1.y = patch[nlo][ii][2 * half + 5];
            v2f wa0, wa1;                    // A frag = W^T: lane -> d
            if constexpr (PACKED) {
                float4 wv = wpbase[ii * (OC * 32)];   // one global_load_b128
                wa0.x = wv.x; wa0.y = wv.y; wa1.x = wv.z; wa1.y = wv.w;
            } else {
                const float* wb = wbase + ii * wstride;
                wa0.x = wb[(2 * half + 0) * DV];
                wa0.y = wb[(2 * half + 1) * DV];
                wa1.x = wb[(2 * half + 4) * DV];
                wa1.y = wb[(2 * half + 5) * DV];
            }
            v8f vt = vz;
            vt = __builtin_amdgcn_wmma_f32_16x16x4_f32(false, wa0, false, pb0,
                                                       (short)0, vt, false, false);
            vt = __builtin_amdgcn_wmma_f32_16x16x4_f32(false, wa1, false, pb1,
                                                       (short)0, vt, false, false);
            float cf = cL[ii][o][nlo];       // one scalar coefficient per lane
            s += cf * vt;
        }

        // ---- Phase C: squash. 8 in-register FMAs + ONE xor-16 shuffle.
        float ss = 0.f;
        #pragma unroll
        for (int j = 0; j < 8; ++j) ss = fmaf(s[j], s[j], ss);
        ss += __shfl_xor(ss, 16, 32);
        float scale = (ss / (1.f + ss)) * __frsqrt_rn(ss + CAPS_EPS);
        outf = scale * s;

        if (r == NROUT - 1) {
            // Final store: lane owns pos, elements are 8 contiguous d -> b128s
            float* dst = out + ((pbase + nlo) * OC + o) * DV + 8 * half;
            *(float4*)(dst)     = make_float4(outf[0], outf[1], outf[2], outf[3]);
            *(float4*)(dst + 4) = make_float4(outf[4], outf[5], outf[6], outf[7]);
        } else {
            // ---- Phase D: b[pos,i,o] += sum_d votes*out (in-register dot +
            // one xor-16 shuffle + single-writer ds_add_f32).
            for (int ii = 0; ii < NI; ++ii) {
                v2f pb0, pb1;
                pb0.x = patch[nlo][ii][2 * half + 0];
                pb0.y = patch[nlo][ii][2 * half + 1];
                pb1.x = patch[nlo][ii][2 * half + 4];
                pb1.y = patch[nlo][ii][2 * half + 5];
                v2f wa0, wa1;
                if constexpr (PACKED) {
                    float4 wv = wpbase[ii * (OC * 32)];
                    wa0.x = wv.x; wa0.y = wv.y; wa1.x = wv.z; wa1.y = wv.w;
                } else {
                    const float* wb = wbase + ii * wstride;
                    wa0.x = wb[(2 * half + 0) * DV];
                    wa0.y = wb[(2 * half + 1) * DV];
                    wa1.x = wb[(2 * half + 4) * DV];
                    wa1.y = wb[(2 * half + 5) * DV];
                }
                v8f vt = vz;
                vt = __builtin_amdgcn_wmma_f32_16x16x4_f32(false, wa0, false, pb0,
                                                           (short)0, vt, false, false);
                vt = __builtin_amdgcn_wmma_f32_16x16x4_f32(false, wa1, false, pb1,
                                                           (short)0, vt, false, false);
                float acc = 0.f;
                #pragma unroll
                for (int j = 0; j < 8; ++j) acc = fmaf(vt[j], outf[j], acc);
                acc += __shfl_xor(acc, 16, 32);
                if (half == 0)
                    atomicAdd(&bL[nlo][ii][o], acc);
            }
        }
        __syncthreads();
    }
}

extern "C" void kernel_launch(void* const* d_in, const int* in_sizes, int n_in,
                              void* d_out, int out_size, void* d_ws, size_t ws_size,
                              hipStream_t stream) {
    const float* x    = (const float*)d_in[0];
    const float* Wg   = (const float*)d_in[1];
    const float* bias = (const float*)d_in[2];
    float* outp = (float*)d_out;
    dim3 grid(NPOS / PPW);   // 242 workgroups, 16 positions each
    dim3 block(NT);          // 16 wave32 waves -> one output capsule per wave
    const size_t wp_bytes = (size_t)WP_ELEMS * sizeof(float4);   // 589824 B
    if (ws_size >= wp_bytes) {
        float4* Wpk = (float4*)d_ws;
        repack_w_kernel<<<(WP_ELEMS + 255) / 256, 256, 0, stream>>>(Wg, Wpk);
        caps_conv_route_kernel<true><<<grid, block, 0, stream>>>(x, Wg, Wpk, bias, outp);
    } else {
        caps_conv_route_kernel<false><<<grid, block, 0, stream>>>(x, Wg, nullptr, bias, outp);
    }
}